// lstm_rnn_gru_91311004713157
// MI455X (gfx1250) — compile-verified
//
#include <hip/hip_runtime.h>

typedef __attribute__((ext_vector_type(2))) float v2f;
typedef __attribute__((ext_vector_type(8))) float v8f;

#define B_TOT 1024
#define T_LEN 4096
#define HID   20
#define NG    80          // 4*HID gate columns
#define ROWS  16          // batch rows per block (one WMMA M-tile)
#define GST   84          // LDS gate-row stride (avoids bank aliasing: 8*84 % 64 == 32)

__device__ __forceinline__ float sigm(float x) {
    return 1.0f / (1.0f + __expf(-x));
}
__device__ __forceinline__ float tanh_f(float x) {
    // tanh(x) = 1 - 2/(e^{2x}+1); saturates correctly for large |x|
    return 1.0f - 2.0f / (__expf(2.0f * x) + 1.0f);
}

__global__ void __launch_bounds__(32)
lstm_seq_kernel(const float* __restrict__ xin,
                const float* __restrict__ Wih1,
                const float* __restrict__ Whh1,
                const float* __restrict__ bih1,
                const float* __restrict__ bhh1,
                const float* __restrict__ Wih2,
                const float* __restrict__ Whh2,
                const float* __restrict__ bih2,
                const float* __restrict__ bhh2,
                const int*   __restrict__ futp,
                float* __restrict__ out)
{
    __shared__ float lds_h1[ROWS * HID];   // layer-1 hidden state (16x20)
    __shared__ float lds_c1[ROWS * HID];   // layer-1 cell state  (16x20)
    __shared__ float lds_g [ROWS * GST];   // gate pre-activations (16x80, stride 84)
    __shared__ float lds_x [ROWS];         // per-row input this step
    __shared__ float s_wih[NG];            // W_ih1 (80x1 flattened)
    __shared__ float s_bs [NG];            // b_ih1 + b_hh1
    __shared__ float s_w2 [NG];            // W_ih2 (4x20 flattened)
    __shared__ float s_whh2[4];
    __shared__ float s_b2 [4];             // b_ih2 + b_hh2

    const int lane  = threadIdx.x;
    const int nu    = lane & 15;     // M (A/D) or N (B/D) index within tile
    const int khalf = lane >> 4;     // 0: K=0,1 / rows 0..7 ; 1: K=2,3 / rows 8..15
    const int row0  = blockIdx.x * ROWS;
    const int fut   = *futp;
    const int TT    = T_LEN + fut;

    // ---- init shared weights / states ----
    for (int i = lane; i < NG; i += 32) {
        s_wih[i] = Wih1[i];
        s_bs [i] = bih1[i] + bhh1[i];
        s_w2 [i] = Wih2[i];
    }
    if (lane < 4) {
        s_whh2[lane] = Whh2[lane];
        s_b2 [lane]  = bih2[lane] + bhh2[lane];
    }
    for (int i = lane; i < ROWS * HID; i += 32) {
        lds_h1[i] = 0.0f;
        lds_c1[i] = 0.0f;
    }

    // ---- loop-invariant B fragments of Whh1^T (20x80): B[k][n] = Whh1[n][k]
    //      tile (kc,n): K rows 4kc..4kc+3, N cols 16n..16n+15
    v2f bf[5][5];
#pragma unroll
    for (int kc = 0; kc < 5; ++kc) {
#pragma unroll
        for (int n = 0; n < 5; ++n) {
            const float* p = Whh1 + (16 * n + nu) * HID + 4 * kc + 2 * khalf;
            v2f b; b.x = p[0]; b.y = p[1];
            bf[kc][n] = b;
        }
    }

    // per-lane recurrent state: each lane owns 10 (row,h) pairs of c1
    float c1v[10];
#pragma unroll
    for (int j = 0; j < 10; ++j) c1v[j] = 0.0f;
    float h2v = 0.0f, c2v = 0.0f;

    const int er  = lane >> 1;         // elementwise row (0..15)
    const int eh0 = (lane & 1) * 10;   // first hidden unit this lane owns

    __syncthreads();

    for (int t = 0; t < TT; ++t) {
        // ---- fetch input (main phase); future phase keeps c2 feedback in lds_x
        if (t < T_LEN) {
            if (lane < ROWS)
                lds_x[lane] = xin[(size_t)(row0 + lane) * T_LEN + t];
        }
        __syncthreads();

        // ---- WMMA: gates(16x80) = h1(16x20) @ Whh1^T(20x80), f32, K chunks of 4
        v8f acc[5];
#pragma unroll
        for (int n = 0; n < 5; ++n) {
            v8f z = {0.f, 0.f, 0.f, 0.f, 0.f, 0.f, 0.f, 0.f};
            acc[n] = z;
        }
#pragma unroll
        for (int kc = 0; kc < 5; ++kc) {
            v2f a;
            const float* hp = &lds_h1[nu * HID + 4 * kc + 2 * khalf];
            a.x = hp[0]; a.y = hp[1];
#pragma unroll
            for (int n = 0; n < 5; ++n) {
                acc[n] = __builtin_amdgcn_wmma_f32_16x16x4_f32(
                    false, a, false, bf[kc][n], (short)0, acc[n], false, false);
            }
        }
        // scatter D tiles -> LDS gate buffer [row][col]
#pragma unroll
        for (int n = 0; n < 5; ++n) {
#pragma unroll
            for (int v = 0; v < 8; ++v) {
                lds_g[(v + 8 * khalf) * GST + 16 * n + nu] = acc[n][v];
            }
        }
        __syncthreads();

        // ---- layer-1 elementwise: add x/bias terms, apply LSTM cell
        {
            const float  xr = lds_x[er];
            const float* gr = &lds_g[er * GST];
#pragma unroll
            for (int j = 0; j < 10; ++j) {
                const int h = eh0 + j;
                float gi = gr[h]      + xr * s_wih[h]      + s_bs[h];
                float gf = gr[20 + h] + xr * s_wih[20 + h] + s_bs[20 + h];
                float gg = gr[40 + h] + xr * s_wih[40 + h] + s_bs[40 + h];
                float go = gr[60 + h] + xr * s_wih[60 + h] + s_bs[60 + h];
                float cn = sigm(gf) * c1v[j] + sigm(gi) * tanh_f(gg);
                c1v[j] = cn;
                float hn = sigm(go) * tanh_f(cn);
                lds_h1[er * HID + h] = hn;
                lds_c1[er * HID + h] = cn;
            }
        }
        __syncthreads();

        // ---- layer-2 LSTM cell (input = c1), one lane per batch row
        if (lane < ROWS) {
            float g2[4];
#pragma unroll
            for (int g = 0; g < 4; ++g) g2[g] = s_b2[g] + h2v * s_whh2[g];
            const float* cr = &lds_c1[lane * HID];
#pragma unroll
            for (int h = 0; h < HID; ++h) {
                float cv = cr[h];
#pragma unroll
                for (int g = 0; g < 4; ++g) g2[g] += cv * s_w2[g * HID + h];
            }
            float cn2 = sigm(g2[1]) * c2v + sigm(g2[0]) * tanh_f(g2[2]);
            c2v = cn2;
            h2v = sigm(g2[3]) * tanh_f(cn2);
            out[(size_t)(row0 + lane) * TT + t] = cn2;
            lds_x[lane] = cn2;   // feedback input for the future phase
        }
        __syncthreads();
    }
}

extern "C" void kernel_launch(void* const* d_in, const int* in_sizes, int n_in,
                              void* d_out, int out_size, void* d_ws, size_t ws_size,
                              hipStream_t stream)
{
    const float* xin  = (const float*)d_in[0];
    const float* Wih1 = (const float*)d_in[1];
    const float* Whh1 = (const float*)d_in[2];
    const float* bih1 = (const float*)d_in[3];
    const float* bhh1 = (const float*)d_in[4];
    const float* Wih2 = (const float*)d_in[5];
    const float* Whh2 = (const float*)d_in[6];
    const float* bih2 = (const float*)d_in[7];
    const float* bhh2 = (const float*)d_in[8];
    const int*   futp = (const int*)d_in[9];
    float* out = (float*)d_out;

    dim3 grid(B_TOT / ROWS);   // 64 blocks x 16 batch rows
    dim3 block(32);            // one wave32 per block
    lstm_seq_kernel<<<grid, block, 0, stream>>>(xin, Wih1, Whh1, bih1, bhh1,
                                                Wih2, Whh2, bih2, bhh2, futp, out);
}